// TSMNetLieBN_35519379538485
// MI455X (gfx1250) — compile-verified
//
#include <hip/hip_runtime.h>
#include <hip/hip_bf16.h>
#include <math.h>

// ---- problem constants (from reference) ----
#define BATCH 256
#define NCH   32
#define TLEN  2000
#define NTF   4
#define NSF   40
#define NSUB  20
#define TK    25
#define NCLS  4
#define NTAN  210            // NSUB*(NSUB+1)/2
#define REIG_EPS 1e-4f

// ---- tiling ----
#define NT    80             // time tile (2000 = 25 * 80), multiple of 16
#define NTILES 25
#define XW    (NT + TK - 1)  // 104 (12 halo each side)
#define PADL  12
#define KC    (NTF * NCH)    // 128 contraction dim for conv2 GEMM
#define MPAD  48             // NSF=40 padded to 3x16
#define XELEMS (NCH * XW)    // 3328 = 256 * 13 (no tail)

typedef float v2f __attribute__((ext_vector_type(2)));
typedef float v8f __attribute__((ext_vector_type(8)));

// D = A(16x4) * B(4x16) + C(16x16), fp32 WMMA (verified lowering round 1)
static __device__ __forceinline__ v8f wmma4(v2f a, v2f b, v8f c) {
    return __builtin_amdgcn_wmma_f32_16x16x4_f32(
        false, a, false, b, (short)0, c, false, false);
}

// Async DMA: global -> LDS, 4 bytes per lane, tracked by ASYNCcnt.
// VDST = LDS byte offset (generic LDS pointers carry the AS3 offset in the
// low 32 bits), VADDR = 64-bit global address (VGPR pair), per ISA 15.18.3.
static __device__ __forceinline__ void async_copy_b32(void* lds_dst, const float* gsrc) {
    unsigned lds_off = (unsigned)(size_t)lds_dst;
    asm volatile("global_load_async_to_lds_b32 %0, %1, off"
                 :: "v"(lds_off), "v"(gsrc) : "memory");
}
static __device__ __forceinline__ void wait_async0() {
#if __has_builtin(__builtin_amdgcn_s_wait_asynccnt)
    __builtin_amdgcn_s_wait_asynccnt(0);
#else
    asm volatile("s_wait_asynccnt 0x0" ::: "memory");
#endif
}

// =====================================================================
// Kernel A: per-batch fused conv1 + conv2(WMMA) + covariance(WMMA)
// grid = BATCH blocks, 256 threads (8 waves)
// =====================================================================
__global__ __launch_bounds__(256)
void conv_cov_kernel(const float* __restrict__ x,
                     const float* __restrict__ w1, const float* __restrict__ b1,
                     const float* __restrict__ w2, const float* __restrict__ b2,
                     float* __restrict__ Cout) {
    __shared__ float sW2[MPAD][KC];      // conv2 weights, rows 40..47 zero
    __shared__ float sW1[NTF][TK];
    __shared__ float sB1[NTF];
    __shared__ float sB2[MPAD];
    __shared__ float sX[2][NCH][XW];     // double-buffered x tile (async DMA dest)
    __shared__ float sH1[KC][NT];        // conv1 output tile (K-major GEMM B)
    __shared__ float sH2[MPAD][NT];      // conv2 output tile
    __shared__ float sSum[NSF];
    __shared__ float sSumPart[NSF][4];

    const int b    = blockIdx.x;
    const int tid  = threadIdx.x;
    const int lane = tid & 31;
    const int wave = tid >> 5;

    const float* xb = x + (size_t)b * NCH * TLEN;

    // ---- prologue: kick off async DMA of tile 0 into buffer 0 ----
    {
        #pragma unroll
        for (int it = 0; it < XELEMS / 256; ++it) {
            int i = tid + it * 256;
            int c = i / XW, j = i % XW;
            int tg = -PADL + j;                   // t0 = 0
            if (tg < 0) tg = -tg;                 // reflect (left edge only)
            async_copy_b32(&sX[0][c][j], xb + c * TLEN + tg);
        }
    }

    // ---- stage weights ----
    for (int i = tid; i < MPAD * KC; i += 256) {
        int s = i / KC, k = i % KC;
        sW2[s][k] = (s < NSF) ? w2[s * KC + k] : 0.0f;
    }
    if (tid < NTF * TK) sW1[tid / TK][tid % TK] = w1[tid];
    if (tid < NTF)      sB1[tid] = b1[tid];
    if (tid < MPAD)     sB2[tid] = (tid < NSF) ? b2[tid] : 0.0f;
    __syncthreads();

    // ---- per-thread conv1 state: fixed (f,c) row, taps in registers ----
    const int row  = tid >> 1;          // 0..127 == f*32 + c
    const int half = tid & 1;           // which 40-wide half of the tile
    const int f1   = row >> 5;
    const int c1   = row & 31;
    float w1r[TK];
    #pragma unroll
    for (int kk = 0; kk < TK; ++kk) w1r[kk] = sW1[f1][kk];
    const float b1f = sB1[f1];

    // persistent Gram accumulators: wave w owns 16x16 tile w of the 3x3
    // grid; wave 0 additionally owns tile (32,32).
    v8f gAcc0 = {};
    v8f gAcc1 = {};
    // persistent feature-sum partial: threads 0..159 -> (feat = tid/4, quarter)
    float sumPart = 0.0f;
    const int sfeat = tid >> 2;         // valid when < NSF
    const int squad = tid & 3;

    const int fr     = lane & 15;       // fragment row/col within 16
    const int khalf  = (lane >> 4) * 2; // K sub-pair select per ISA layout
    const int rowOff = (lane < 16) ? 0 : 8;

    for (int tile = 0; tile < NTILES; ++tile) {
        const int buf = tile & 1;
        // ---- wait for this tile's DMA, make visible to all waves ----
        wait_async0();
        __syncthreads();

        // ---- conv1: register-resident FIR, sliding 32-wide window ----
        {
            const int tb = half * 40;
            for (int ch = 0; ch < 5; ++ch) {
                const int xoff = tb + ch * 8;
                float xwin[32];
                #pragma unroll
                for (int j = 0; j < 32; ++j) xwin[j] = sX[buf][c1][xoff + j];
                #pragma unroll
                for (int j = 0; j < 8; ++j) {
                    float acc = b1f;
                    #pragma unroll
                    for (int kk = 0; kk < TK; ++kk)
                        acc = fmaf(w1r[kk], xwin[j + kk], acc);
                    sH1[row][xoff + j] = acc;
                }
            }
        }
        __syncthreads();

        // ---- prefetch next tile into the other buffer (overlaps WMMA) ----
        if (tile + 1 < NTILES) {
            const int t0n = (tile + 1) * NT;
            #pragma unroll
            for (int it = 0; it < XELEMS / 256; ++it) {
                int i = tid + it * 256;
                int c = i / XW, j = i % XW;
                int tg = t0n - PADL + j;
                if (tg >= TLEN) tg = 2 * TLEN - 2 - tg;   // reflect (right edge)
                async_copy_b32(&sX[buf ^ 1][c][j], xb + c * TLEN + tg);
            }
        }

        // ---- conv2 GEMM: (48x128) @ (128x80) via WMMA f32 16x16x4 ----
        for (int rep = 0; rep < 2; ++rep) {
            int tIdx = wave + rep * 8;
            if (tIdx < 15) {
                const int m0 = (tIdx / 5) * 16;
                const int n0 = (tIdx % 5) * 16;
                v8f acc = {};
                #pragma unroll
                for (int k0 = 0; k0 < KC; k0 += 4) {
                    v2f a, bb;
                    a.x  = sW2[m0 + fr][k0 + khalf];
                    a.y  = sW2[m0 + fr][k0 + khalf + 1];
                    bb.x = sH1[k0 + khalf][n0 + fr];
                    bb.y = sH1[k0 + khalf + 1][n0 + fr];
                    acc = wmma4(a, bb, acc);
                }
                const int n = n0 + fr;
                #pragma unroll
                for (int r = 0; r < 8; ++r) {
                    int m = m0 + r + rowOff;
                    sH2[m][n] = acc[r] + sB2[m];
                }
            }
        }
        __syncthreads();

        // ---- Gram accumulate: G += H2 * H2^T, K = NT ----
        {
            const int m0 = (wave / 3) * 16;
            const int n0 = (wave % 3) * 16;
            #pragma unroll
            for (int k0 = 0; k0 < NT; k0 += 4) {
                v2f a, bb;
                a.x  = sH2[m0 + fr][k0 + khalf];
                a.y  = sH2[m0 + fr][k0 + khalf + 1];
                bb.x = sH2[n0 + fr][k0 + khalf];
                bb.y = sH2[n0 + fr][k0 + khalf + 1];
                gAcc0 = wmma4(a, bb, gAcc0);
            }
            if (wave == 0) {   // extra diagonal tile (32,32)
                #pragma unroll
                for (int k0 = 0; k0 < NT; k0 += 4) {
                    v2f a, bb;
                    a.x  = sH2[32 + fr][k0 + khalf];
                    a.y  = sH2[32 + fr][k0 + khalf + 1];
                    bb.x = bb.x = sH2[32 + fr][k0 + khalf];
                    bb.y = sH2[32 + fr][k0 + khalf + 1];
                    gAcc1 = wmma4(a, bb, gAcc1);
                }
            }
        }
        // ---- running feature sums (4 threads per feature) ----
        if (sfeat < NSF) {
            float s = 0.0f;
            #pragma unroll 4
            for (int t = squad * 20; t < squad * 20 + 20; ++t)
                s += sH2[sfeat][t];
            sumPart += s;
        }
    }

    __syncthreads();
    // reuse sH1 storage for the 48x48 Gram write-out
    float (*sG)[MPAD] = (float(*)[MPAD])(&sH1[0][0]);
    {
        const int m0 = (wave / 3) * 16;
        const int n0 = (wave % 3) * 16;
        #pragma unroll
        for (int r = 0; r < 8; ++r)
            sG[m0 + r + rowOff][n0 + fr] = gAcc0[r];
        if (wave == 0) {
            #pragma unroll
            for (int r = 0; r < 8; ++r)
                sG[32 + r + rowOff][32 + fr] = gAcc1[r];
        }
    }
    if (sfeat < NSF) sSumPart[sfeat][squad] = sumPart;
    __syncthreads();
    if (tid < NSF)
        sSum[tid] = sSumPart[tid][0] + sSumPart[tid][1] +
                    sSumPart[tid][2] + sSumPart[tid][3];
    __syncthreads();
    // C = (G - s_i s_j / T) / (T-1)
    const float invT   = 1.0f / (float)TLEN;
    const float invTm1 = 1.0f / (float)(TLEN - 1);
    for (int i = tid; i < NSF * NSF; i += 256) {
        int ii = i / NSF, jj = i % NSF;
        float g = sG[ii][jj] - sSum[ii] * sSum[jj] * invT;
        Cout[(size_t)b * (NSF * NSF) + i] = g * invTm1;
    }
}

// =====================================================================
// Kernel B: bimap + Jacobi eigh + matrix-log + triu + classifier
// grid = BATCH blocks, 32 threads (one wave per 20x20 SPD matrix).
// eigh(U max(w,eps) U^T) == (max(w,eps), U), so the reference's second
// eigh is the identity; L = U log(max(w,eps)) U^T from one decomposition.
// =====================================================================
__global__ __launch_bounds__(32)
void spd_head_kernel(const float* __restrict__ Cin,
                     const float* __restrict__ Wb,
                     const float* __restrict__ clfw,
                     const float* __restrict__ clfb,
                     float* __restrict__ out) {
    __shared__ float sC[NSF][NSF];
    __shared__ float sW[NSUB][NSF];
    __shared__ float sM[NSUB][NSF];
    __shared__ float sS[NSUB][NSUB];
    __shared__ float sU[NSUB][NSUB];
    __shared__ float sLW[NSUB];
    __shared__ float sRed[NCLS][32];

    const int b = blockIdx.x;
    const int tid = threadIdx.x;

    for (int i = tid; i < NSF * NSF; i += 32)
        sC[i / NSF][i % NSF] = Cin[(size_t)b * (NSF * NSF) + i];
    for (int i = tid; i < NSUB * NSF; i += 32)
        sW[i / NSF][i % NSF] = Wb[i];
    __syncthreads();
    // M = W * C
    for (int i = tid; i < NSUB * NSF; i += 32) {
        int r = i / NSF, c = i % NSF;
        float acc = 0.0f;
        for (int k = 0; k < NSF; ++k) acc += sW[r][k] * sC[k][c];
        sM[r][c] = acc;
    }
    __syncthreads();
    // S = M * W^T ; U = I
    for (int i = tid; i < NSUB * NSUB; i += 32) {
        int r = i / NSUB, c = i % NSUB;
        float acc = 0.0f;
        for (int k = 0; k < NSF; ++k) acc += sM[r][k] * sW[c][k];
        sS[r][c] = acc;
        sU[r][c] = (r == c) ? 1.0f : 0.0f;
    }
    __syncthreads();

    // cyclic Jacobi sweeps (branches are wave-uniform: same LDS scalars)
    for (int sweep = 0; sweep < 10; ++sweep) {
        for (int p = 0; p < NSUB - 1; ++p) {
            for (int q = p + 1; q < NSUB; ++q) {
                float apq = sS[p][q];
                float app = sS[p][p];
                float aqq = sS[q][q];
                if (fabsf(apq) <= 1e-12f) continue;
                float tau = (aqq - app) / (2.0f * apq);
                float tt  = ((tau >= 0.0f) ? 1.0f : -1.0f) /
                            (fabsf(tau) + sqrtf(1.0f + tau * tau));
                float c = 1.0f / sqrtf(1.0f + tt * tt);
                float s = tt * c;

                int j = tid;
                float rp = 0.f, rq = 0.f, up = 0.f, uq = 0.f;
                if (j < NSUB) {
                    rp = sS[p][j]; rq = sS[q][j];
                    up = sU[j][p]; uq = sU[j][q];
                }
                __syncthreads();
                if (j < NSUB) {
                    float np = c * rp - s * rq;
                    float nq = s * rp + c * rq;
                    if (j != p && j != q) {
                        sS[p][j] = np; sS[j][p] = np;
                        sS[q][j] = nq; sS[j][q] = nq;
                    }
                    sU[j][p] = c * up - s * uq;
                    sU[j][q] = s * up + c * uq;
                }
                if (tid == 0) {
                    sS[p][p] = app - tt * apq;
                    sS[q][q] = aqq + tt * apq;
                    sS[p][q] = 0.0f; sS[q][p] = 0.0f;
                }
                __syncthreads();
            }
        }
    }

    if (tid < NSUB) sLW[tid] = logf(fmaxf(sS[tid][tid], REIG_EPS));
    __syncthreads();

    // z = scaled triu(L), out = z @ clf_w^T + clf_b
    float part[NCLS] = {0.f, 0.f, 0.f, 0.f};
    const float SQRT2 = 1.4142135623730951f;
    for (int idx = tid; idx < NTAN; idx += 32) {
        int i = 0, rem = idx;
        while (rem >= (NSUB - i)) { rem -= (NSUB - i); ++i; }
        int j = i + rem;
        float L = 0.0f;
        for (int k = 0; k < NSUB; ++k) L += sU[i][k] * sLW[k] * sU[j][k];
        float z = L * ((i == j) ? 1.0f : SQRT2);
        #pragma unroll
        for (int cc = 0; cc < NCLS; ++cc)
            part[cc] += clfw[cc * NTAN + idx] * z;
    }
    #pragma unroll
    for (int cc = 0; cc < NCLS; ++cc) sRed[cc][tid] = part[cc];
    __syncthreads();
    if (tid < NCLS) {
        float acc = clfb[tid];
        for (int l = 0; l < 32; ++l) acc += sRed[tid][l];
        out[(size_t)b * NCLS + tid] = acc;
    }
}

// =====================================================================
extern "C" void kernel_launch(void* const* d_in, const int* in_sizes, int n_in,
                              void* d_out, int out_size, void* d_ws, size_t ws_size,
                              hipStream_t stream) {
    (void)in_sizes; (void)n_in; (void)out_size; (void)ws_size;
    const float* x   = (const float*)d_in[0];
    const float* w1  = (const float*)d_in[1];
    const float* b1  = (const float*)d_in[2];
    const float* w2  = (const float*)d_in[3];
    const float* b2  = (const float*)d_in[4];
    const float* Wb  = (const float*)d_in[5];
    const float* cw  = (const float*)d_in[6];
    const float* cb  = (const float*)d_in[7];
    float* out = (float*)d_out;
    float* Cws = (float*)d_ws;    // BATCH * 40 * 40 floats = 1.64 MB

    conv_cov_kernel<<<BATCH, 256, 0, stream>>>(x, w1, b1, w2, b2, Cws);
    spd_head_kernel<<<BATCH, 32, 0, stream>>>(Cws, Wb, cw, cb, out);
}